// GNNEncoder_85933705658442
// MI455X (gfx1250) — compile-verified
//
#include <hip/hip_runtime.h>
#include <hip/hip_bf16.h>

#define DIN   26
#define KPAD  32
#define DH    128

typedef __attribute__((ext_vector_type(2))) float v2f;
typedef __attribute__((ext_vector_type(8))) float v8f;

// ---------------------------------------------------------------------------
// small utility kernels
// ---------------------------------------------------------------------------
__global__ void zero_kernel(float* __restrict__ p, int n) {
    int i = blockIdx.x * blockDim.x + threadIdx.x;
    if (i < n) p[i] = 0.0f;
}

__global__ void pad_x_kernel(const float* __restrict__ x, float* __restrict__ xp, int N) {
    int i = blockIdx.x * blockDim.x + threadIdx.x;     // over N*KPAD
    if (i >= N * KPAD) return;
    int n = i >> 5, c = i & (KPAD - 1);
    xp[i] = (c < DIN) ? x[n * DIN + c] : 0.0f;
}

__global__ void pad_w1_kernel(const float* __restrict__ w1, float* __restrict__ w1p) {
    int i = blockIdx.x * blockDim.x + threadIdx.x;     // over KPAD*DH
    if (i >= KPAD * DH) return;
    int r = i >> 7;
    w1p[i] = (r < DIN) ? w1[r * DH + (i & (DH - 1))] : 0.0f;
}

__global__ void deg_init_kernel(float* __restrict__ deg, int N) {
    int i = blockIdx.x * blockDim.x + threadIdx.x;
    if (i < N) deg[i] = 1.0f;                          // self-loop contribution
}

__global__ void deg_accum_kernel(const int* __restrict__ ei, float* __restrict__ deg, int E) {
    int e = blockIdx.x * blockDim.x + threadIdx.x;
    if (e < E) unsafeAtomicAdd(&deg[ei[E + e]], 1.0f); // dst = ei[1][e]
}

__global__ void deg_to_dinv_kernel(float* __restrict__ deg, int N) {
    int i = blockIdx.x * blockDim.x + threadIdx.x;
    if (i < N) deg[i] = rsqrtf(deg[i]);                // deg >= 1 always
}

// ---------------------------------------------------------------------------
// WMMA fp32 GEMM:  T[N x 128] = A[N x K] * W[K x 128]
// one block = 16 rows x 128 cols, 8 waves, each wave one 16x16 tile.
// A tile staged in LDS (row stride K+1 to avoid bank conflicts).
// ---------------------------------------------------------------------------
__global__ __launch_bounds__(256)
void gemm_wmma_kernel(const float* __restrict__ A, const float* __restrict__ W,
                      float* __restrict__ T, int K, int kshift) {
    __shared__ float As[16 * (DH + 1)];                // big enough for K<=128
    const int tid    = threadIdx.x;
    const int lane   = tid & 31;
    const int wave   = tid >> 5;                       // 0..7 -> column tile
    const int rowTop = blockIdx.x * 16;
    const int lds_ld = K + 1;

    // cooperative load of 16 x K A-tile
    for (int i = tid; i < 16 * K; i += 256) {
        int r = i >> kshift;
        int c = i & (K - 1);
        As[r * lds_ld + c] = A[(size_t)(rowTop + r) * K + c];
    }
    __syncthreads();

    const int m  = lane & 15;
    const int kb = (lane >> 4) << 1;                   // 0 or 2
    const int col = wave * 16 + m;

    v8f acc = {};
    for (int k = 0; k < K; k += 4) {
        v2f a, b;
        a.x = As[m * lds_ld + k + kb];
        a.y = As[m * lds_ld + k + kb + 1];
        b.x = W[(size_t)(k + kb) * DH + col];
        b.y = W[(size_t)(k + kb + 1) * DH + col];
        acc = __builtin_amdgcn_wmma_f32_16x16x4_f32(false, a, false, b,
                                                    (short)0, acc, false, false);
    }

    // D layout: vgpr v -> row (v + 8*(lane>=16)), col = lane&15
    const int nOut = wave * 16 + (lane & 15);
    const int rOff = (lane >> 4) << 3;
#pragma unroll
    for (int v = 0; v < 8; ++v) {
        T[(size_t)(rowTop + v + rOff) * DH + nOut] = acc[v];
    }
}

// ---------------------------------------------------------------------------
// aggregation
// ---------------------------------------------------------------------------
__global__ void init_self_kernel(const float* __restrict__ T, const float* __restrict__ dinv,
                                 float* __restrict__ Agg, int n) {
    int i = blockIdx.x * blockDim.x + threadIdx.x;     // over N*128
    if (i >= n) return;
    float dv = dinv[i >> 7];
    Agg[i] = T[i] * dv * dv;                           // self-loop: norm = 1/deg
}

// one wave per edge: 32 lanes x float4 = 128 features
__global__ __launch_bounds__(256)
void edge_scatter_kernel(const int* __restrict__ ei, const float* __restrict__ dinv,
                         const float* __restrict__ T, float* __restrict__ Agg, int E) {
    int wid  = (blockIdx.x * blockDim.x + threadIdx.x) >> 5;
    int lane = threadIdx.x & 31;
    if (wid >= E) return;
    int s = ei[wid];
    int d = ei[E + wid];
    float nrm = dinv[s] * dinv[d];
    const float4 v = ((const float4*)(T + (size_t)s * DH))[lane];
    float* dst = Agg + (size_t)d * DH + lane * 4;
    unsafeAtomicAdd(dst + 0, v.x * nrm);
    unsafeAtomicAdd(dst + 1, v.y * nrm);
    unsafeAtomicAdd(dst + 2, v.z * nrm);
    unsafeAtomicAdd(dst + 3, v.w * nrm);
}

__global__ void bias_act_kernel(float* __restrict__ H, const float* __restrict__ b,
                                int n, int do_relu) {
    int i = blockIdx.x * blockDim.x + threadIdx.x;     // over N*128
    if (i >= n) return;
    float v = H[i] + b[i & (DH - 1)];
    H[i] = do_relu ? fmaxf(v, 0.0f) : v;
}

// ---------------------------------------------------------------------------
// global mean pool
// ---------------------------------------------------------------------------
__global__ void pool_accum_kernel(const float* __restrict__ H, const int* __restrict__ batch,
                                  float* __restrict__ out, int n) {
    int i = blockIdx.x * blockDim.x + threadIdx.x;     // over N*128
    if (i >= n) return;
    int g = batch[i >> 7];
    unsafeAtomicAdd(&out[(size_t)g * DH + (i & (DH - 1))], H[i]);
}

__global__ void count_accum_kernel(const int* __restrict__ batch, float* __restrict__ cnt, int N) {
    int i = blockIdx.x * blockDim.x + threadIdx.x;
    if (i < N) unsafeAtomicAdd(&cnt[batch[i]], 1.0f);
}

__global__ void pool_div_kernel(float* __restrict__ out, const float* __restrict__ cnt, int n) {
    int i = blockIdx.x * blockDim.x + threadIdx.x;     // over G*128
    if (i >= n) return;
    out[i] /= fmaxf(cnt[i >> 7], 1.0f);
}

// ---------------------------------------------------------------------------
// launch
// ---------------------------------------------------------------------------
static inline size_t align_up(size_t v, size_t a) { return (v + a - 1) & ~(a - 1); }

extern "C" void kernel_launch(void* const* d_in, const int* in_sizes, int n_in,
                              void* d_out, int out_size, void* d_ws, size_t ws_size,
                              hipStream_t stream) {
    const float* x    = (const float*)d_in[0];
    const int*   ei   = (const int*)  d_in[1];
    const int*   bat  = (const int*)  d_in[2];
    const float* W1   = (const float*)d_in[3];
    const float* b1   = (const float*)d_in[4];
    const float* W2   = (const float*)d_in[5];
    const float* b2   = (const float*)d_in[6];
    const float* W3   = (const float*)d_in[7];
    const float* b3   = (const float*)d_in[8];
    float* out = (float*)d_out;

    const int N = in_sizes[0] / DIN;       // 100000
    const int E = in_sizes[1] / 2;         // 1600000
    const int G = out_size / DH;           // 2048

    // workspace carve-up
    char* base = (char*)d_ws;
    size_t off = 0;
    float* dinv = (float*)(base + off); off = align_up(off + (size_t)N * 4, 256);
    float* xpad = (float*)(base + off); off = align_up(off + (size_t)N * KPAD * 4, 256);
    float* w1p  = (float*)(base + off); off = align_up(off + (size_t)KPAD * DH * 4, 256);
    float* T    = (float*)(base + off); off = align_up(off + (size_t)N * DH * 4, 256);
    float* Agg  = (float*)(base + off); off = align_up(off + (size_t)N * DH * 4, 256);
    float* cnt  = (float*)(base + off); off = align_up(off + (size_t)G * 4, 256);
    (void)ws_size; (void)n_in;

    const int BT = 256;
    const int nd  = N * DH;                // 12.8M elements
    const int gNd = (nd + BT - 1) / BT;
    const int gN  = (N + BT - 1) / BT;
    const int gE  = (E + BT - 1) / BT;

    // --- preprocessing -----------------------------------------------------
    pad_x_kernel <<<(N * KPAD + BT - 1) / BT, BT, 0, stream>>>(x, xpad, N);
    pad_w1_kernel<<<(KPAD * DH + BT - 1) / BT, BT, 0, stream>>>(W1, w1p);
    deg_init_kernel<<<gN, BT, 0, stream>>>(dinv, N);
    deg_accum_kernel<<<gE, BT, 0, stream>>>(ei, dinv, E);
    deg_to_dinv_kernel<<<gN, BT, 0, stream>>>(dinv, N);

    const int gemmBlocks = N / 16;                     // N multiple of 16
    const int edgeBlocks = (E * 32 + BT - 1) / BT;     // one wave per edge

    // --- layer 1 (K = 32, padded) -----------------------------------------
    gemm_wmma_kernel<<<gemmBlocks, BT, 0, stream>>>(xpad, w1p, T, KPAD, 5);
    init_self_kernel<<<gNd, BT, 0, stream>>>(T, dinv, Agg, nd);
    edge_scatter_kernel<<<edgeBlocks, BT, 0, stream>>>(ei, dinv, T, Agg, E);
    bias_act_kernel<<<gNd, BT, 0, stream>>>(Agg, b1, nd, 1);

    // --- layer 2 (K = 128) -------------------------------------------------
    gemm_wmma_kernel<<<gemmBlocks, BT, 0, stream>>>(Agg, W2, T, DH, 7);
    init_self_kernel<<<gNd, BT, 0, stream>>>(T, dinv, Agg, nd);
    edge_scatter_kernel<<<edgeBlocks, BT, 0, stream>>>(ei, dinv, T, Agg, E);
    bias_act_kernel<<<gNd, BT, 0, stream>>>(Agg, b2, nd, 1);

    // --- layer 3 (K = 128, no relu) ---------------------------------------
    gemm_wmma_kernel<<<gemmBlocks, BT, 0, stream>>>(Agg, W3, T, DH, 7);
    init_self_kernel<<<gNd, BT, 0, stream>>>(T, dinv, Agg, nd);
    edge_scatter_kernel<<<edgeBlocks, BT, 0, stream>>>(ei, dinv, T, Agg, E);
    bias_act_kernel<<<gNd, BT, 0, stream>>>(Agg, b3, nd, 0);

    // --- global mean pool --------------------------------------------------
    zero_kernel<<<(out_size + BT - 1) / BT, BT, 0, stream>>>(out, out_size);
    zero_kernel<<<(G + BT - 1) / BT, BT, 0, stream>>>(cnt, G);
    pool_accum_kernel<<<gNd, BT, 0, stream>>>(Agg, bat, out, nd);
    count_accum_kernel<<<gN, BT, 0, stream>>>(bat, cnt, N);
    pool_div_kernel<<<(out_size + BT - 1) / BT, BT, 0, stream>>>(out, cnt, out_size);
}